// SeparableFiberBundleConv_85667417686336
// MI455X (gfx1250) — compile-verified
//
#include <hip/hip_runtime.h>

typedef __attribute__((ext_vector_type(2))) float v2f;
typedef __attribute__((ext_vector_type(8))) float v8f;

#define NN 10000
#define EE 100000
#define OO 12
#define CC 128
#define KK 32
#define AA 16
#define CA 144   // C + A

// D = A(16x4, f32) * B(4x16, f32) + C(16x16, f32)
__device__ __forceinline__ v8f wmma4(v2f a, v2f b, v8f c) {
  return __builtin_amdgcn_wmma_f32_16x16x4_f32(
      /*neg_a=*/false, a, /*neg_b=*/false, b,
      /*c_mod=*/(short)0, c, /*reuse_a=*/false, /*reuse_b=*/false);
}

__global__ void k_zero(float* __restrict__ p, int n) {
  int i = blockIdx.x * blockDim.x + threadIdx.x;
  int stride = gridDim.x * blockDim.x;
  for (; i < n; i += stride) p[i] = 0.0f;
}

// y[N*O, C] = x[N*O, C] @ Wx^T   (Wx = W_pre[:, :C], row-major [C, C+A])
// 320 threads = 10 waves/block; 750 blocks * 10 = 7500 tiles exactly.
// B fragments (lane-swizzled) staged in LDS once per block: 8 nt * 32 ks * 32 lanes.
__global__ void __launch_bounds__(320)
k_node_pre(const float* __restrict__ x,
           const float* __restrict__ Wpre,
           float* __restrict__ y) {
  __shared__ v2f Bs[8 * 32 * 32];          // 64 KB
  const int tid = threadIdx.x;
  for (int idx = tid; idx < 8 * 32 * 32; idx += 320) {
    int ln = idx & 31;
    int ks = (idx >> 5) & 31;
    int nt = idx >> 10;
    int n  = nt * 16 + (ln & 15);
    int kA = ks * 4 + 2 * (ln >> 4);
    Bs[idx] = *(const v2f*)(Wpre + (size_t)n * CA + kA);   // B[k][n] = Wpre[n*CA+k]
  }
  __syncthreads();

  const int wave   = blockIdx.x * 10 + (tid >> 5);   // < 7500 always
  const int lane   = tid & 31;
  const int laneM  = lane & 15;
  const int laneHi = lane >> 4;
  const int row0   = wave * 16;

  // A fragments: a[ks].x = A[M][4*ks + 2*laneHi], .y = +1  (ISA 32-bit A layout)
  v2f a[32];
  const float* arow = x + (size_t)(row0 + laneM) * CC + 2 * laneHi;
#pragma unroll
  for (int ks = 0; ks < 32; ++ks) a[ks] = *(const v2f*)(arow + ks * 4);

  for (int nt = 0; nt < 8; ++nt) {
    const v2f* bs = &Bs[(nt * 32) * 32 + lane];
    v8f c = {};
#pragma unroll
    for (int ks = 0; ks < 32; ++ks)
      c = wmma4(a[ks], bs[ks * 32], c);
    const int n = nt * 16 + laneM;
#pragma unroll
    for (int v = 0; v < 8; ++v)
      y[(size_t)(row0 + v + laneHi * 8) * CC + n] = c[v];
  }
}

// Per edge-row tile (16 rows of E*O):
//   kmod = kernel_basis_tile(16x32) @ Wk^T(32x128)
//   eap  = edge_attr_tile(16x16)    @ Wa^T(16x128)   (Wa = W_pre[:, C:])
//   msg  = (y[src,o,:] + eap) * kmod ; hw-atomic fadd into x1[dst,o,:]
// 256 threads = 8 waves/block; 9375 blocks * 8 = 75000 tiles exactly.
__global__ void __launch_bounds__(256)
k_edge_msg(const float* __restrict__ kb,
           const float* __restrict__ ea,
           const int*   __restrict__ eidx,
           const float* __restrict__ Wpre,
           const float* __restrict__ Wker,
           const float* __restrict__ y,
           float* __restrict__ x1) {
  __shared__ v2f BK[8 * 8 * 32];   // 16 KB: W_kernel^T fragments
  __shared__ v2f BE[8 * 4 * 32];   //  8 KB: W_a^T fragments
  const int tid = threadIdx.x;
  for (int idx = tid; idx < 8 * 8 * 32; idx += 256) {
    int ln = idx & 31;
    int ks = (idx >> 5) & 7;
    int nt = idx >> 8;
    int n  = nt * 16 + (ln & 15);
    int kA = ks * 4 + 2 * (ln >> 4);
    BK[idx] = *(const v2f*)(Wker + (size_t)n * KK + kA);
  }
  for (int idx = tid; idx < 8 * 4 * 32; idx += 256) {
    int ln = idx & 31;
    int ks = (idx >> 5) & 3;
    int nt = idx >> 7;
    int n  = nt * 16 + (ln & 15);
    int kA = ks * 4 + 2 * (ln >> 4);
    BE[idx] = *(const v2f*)(Wpre + (size_t)n * CA + CC + kA);
  }
  __syncthreads();

  const int wave   = blockIdx.x * 8 + (tid >> 5);    // < 75000 always
  const int lane   = tid & 31;
  const int laneM  = lane & 15;
  const int laneHi = lane >> 4;
  const int row0   = wave * 16;

  // A fragments
  v2f aK[8], aE[4];
  {
    const float* kr = kb + (size_t)(row0 + laneM) * KK + 2 * laneHi;
#pragma unroll
    for (int ks = 0; ks < 8; ++ks) aK[ks] = *(const v2f*)(kr + ks * 4);
    const float* er = ea + (size_t)(row0 + laneM) * AA + 2 * laneHi;
#pragma unroll
    for (int ks = 0; ks < 4; ++ks) aE[ks] = *(const v2f*)(er + ks * 4);
  }

  // Gather/scatter bases for the 8 output rows this lane touches
  int ybase[8], dbase[8];
#pragma unroll
  for (int v = 0; v < 8; ++v) {
    int r = row0 + v + laneHi * 8;
    int e = r / OO;
    int o = r - e * OO;
    int s = eidx[e];         // src
    int d = eidx[EE + e];    // dst
    ybase[v] = (s * OO + o) * CC;
    dbase[v] = (d * OO + o) * CC;
  }

  for (int nt = 0; nt < 8; ++nt) {
    const int n = nt * 16 + laneM;
    const v2f* bk = &BK[(nt * 8) * 32 + lane];
    const v2f* be = &BE[(nt * 4) * 32 + lane];
    v8f cK = {}, cE = {};
#pragma unroll
    for (int ks = 0; ks < 8; ++ks) cK = wmma4(aK[ks], bk[ks * 32], cK);
#pragma unroll
    for (int ks = 0; ks < 4; ++ks) cE = wmma4(aE[ks], be[ks * 32], cE);
#pragma unroll
    for (int v = 0; v < 8; ++v) {
      float msg = (y[ybase[v] + n] + cE[v]) * cK[v];
      unsafeAtomicAdd(x1 + dbase[v] + n, msg);   // global_atomic_add_f32
    }
  }
}

// fk[p*O+o, c] = sum_k fiber_kernel_basis[p,o,k] * W_fiber[c,k]
__global__ void k_fiber_kernel(const float* __restrict__ fkb,
                               const float* __restrict__ Wf,
                               float* __restrict__ fk) {
  int pq = blockIdx.x;   // 0..143
  int c  = threadIdx.x;  // 0..127
  const float* b = fkb + pq * KK;
  const float* w = Wf + c * KK;
  float acc = 0.f;
#pragma unroll
  for (int k = 0; k < KK; ++k) acc += b[k] * w[k];
  fk[pq * CC + c] = acc;
}

// out[n,p,c] = (1/O) * sum_o x1[n,o,c] * fk[p*O+o, c] + bias[c]
__global__ void k_fiber_conv(const float* __restrict__ x1,
                             const float* __restrict__ fk,
                             const float* __restrict__ bias,
                             float* __restrict__ out) {
  int c = threadIdx.x;   // 0..127
  float bv = bias[c];
  for (int n = blockIdx.x; n < NN; n += gridDim.x) {
    float xv[OO];
#pragma unroll
    for (int o = 0; o < OO; ++o) xv[o] = x1[(size_t)(n * OO + o) * CC + c];
#pragma unroll
    for (int p = 0; p < OO; ++p) {
      float acc = 0.f;
#pragma unroll
      for (int o = 0; o < OO; ++o) acc += xv[o] * fk[(p * OO + o) * CC + c];
      out[(size_t)(n * OO + p) * CC + c] = acc * (1.0f / OO) + bv;
    }
  }
}

extern "C" void kernel_launch(void* const* d_in, const int* in_sizes, int n_in,
                              void* d_out, int out_size, void* d_ws, size_t ws_size,
                              hipStream_t stream) {
  const float* x    = (const float*)d_in[0];
  const float* kb   = (const float*)d_in[1];
  const float* fkb  = (const float*)d_in[2];
  const int*   eidx = (const int*)d_in[3];
  const float* ea   = (const float*)d_in[4];
  const float* Wpre = (const float*)d_in[5];
  const float* Wker = (const float*)d_in[6];
  const float* Wfib = (const float*)d_in[7];
  const float* bias = (const float*)d_in[8];
  float* out = (float*)d_out;

  float* y  = (float*)d_ws;                      // N*O*C floats
  float* x1 = y  + (size_t)NN * OO * CC;         // N*O*C floats
  float* fk = x1 + (size_t)NN * OO * CC;         // 144*128 floats

  k_zero<<<2048, 256, 0, stream>>>(x1, NN * OO * CC);

  // 7500 row-tiles, 10 waves (320 threads) per block -> 750 blocks exactly
  k_node_pre<<<750, 320, 0, stream>>>(x, Wpre, y);

  // 75000 row-tiles, 8 waves per block -> 9375 blocks exactly
  k_edge_msg<<<75000 / 8, 256, 0, stream>>>(kb, ea, eidx, Wpre, Wker, y, x1);

  k_fiber_kernel<<<OO * OO, CC, 0, stream>>>(fkb, Wfib, fk);
  k_fiber_conv<<<1024, CC, 0, stream>>>(x1, fk, bias, out);
}